// DGCNN_cls_53317724012779
// MI455X (gfx1250) — compile-verified
//
#include <hip/hip_runtime.h>
#include <hip/hip_bf16.h>

typedef __attribute__((ext_vector_type(2))) float v2f;
typedef __attribute__((ext_vector_type(8))) float v8f;

#define KNN 20
#define NEG_BIG (-3.0e38f)

// ---------------------------------------------------------------------------
// prep: x (B,3,N) -> xt (B,N,4) zero-padded, plus squared norms sq (B,N)
// ---------------------------------------------------------------------------
__global__ __launch_bounds__(256)
void prep_kernel(const float* __restrict__ x, float* __restrict__ xt,
                 float* __restrict__ sq, int N)
{
    int n = blockIdx.x * 256 + threadIdx.x;
    int b = blockIdx.y;
    size_t p = (size_t)b * N + n;
    float v0 = x[((size_t)b * 3 + 0) * N + n];
    float v1 = x[((size_t)b * 3 + 1) * N + n];
    float v2 = x[((size_t)b * 3 + 2) * N + n];
    float* o = xt + p * 4;
    o[0] = v0; o[1] = v1; o[2] = v2; o[3] = 0.0f;
    sq[p] = v0 * v0 + v1 * v1 + v2 * v2;
}

// ---------------------------------------------------------------------------
// kNN: per batch, Gram matrix tiles via V_WMMA_F32_16X16X4_F32, fused top-20.
// One wave owns 32 rows (two 16-row WMMA M-tiles); lane <-> row for selection.
// Ranking key = 2*inner - ||xj||^2 (row-constant ||xi||^2 is rank-invariant).
// The 2x scale is folded into the A fragments; -||xj||^2 is folded into the
// tile at LDS-write time, so the per-candidate loop is ds_load + compare only.
// Waves in a block are independent (disjoint LDS slices); wave-internal
// LDS store->load needs no block barrier (DS ops complete in order per wave).
// ---------------------------------------------------------------------------
template<int CPAD>
__global__ __launch_bounds__(128)
void knn_kernel(const float* __restrict__ X,   // (B,N,CPAD), pad channels are 0
                const float* __restrict__ sq,  // (B,N)
                int* __restrict__ idx,         // (B,N,KNN)
                int N)
{
    static_assert(CPAD % 4 == 0, "CPAD must be a multiple of 4");
    constexpr int KSTEPS = CPAD / 4;
    const int lane = threadIdx.x & 31;
    const int wave = threadIdx.x >> 5;
    const int b    = blockIdx.y;
    const int i0   = (blockIdx.x * 4 + wave) * 32;      // 32 rows per wave
    const float* Xb  = X  + (size_t)b * N * CPAD;
    const float* sqb = sq + (size_t)b * N;

    __shared__ float tile[4][32][17];                   // +1 pad: conflict-free

    // A fragments (rows fixed for the whole column sweep), scaled by 2.
    // 16x4 f32 A layout: lane&15 = M, (lane>>4)*2 = base K; v2f = {K, K+1}
    v2f a0[KSTEPS], a1[KSTEPS];
    {
        const int m  = lane & 15;
        const int kb = (lane >> 4) * 2;
        const float* r0 = Xb + (size_t)(i0 + m)      * CPAD;
        const float* r1 = Xb + (size_t)(i0 + 16 + m) * CPAD;
        #pragma unroll
        for (int s = 0; s < KSTEPS; ++s) {
            v2f t0 = *(const v2f*)(r0 + s * 4 + kb);
            v2f t1 = *(const v2f*)(r1 + s * 4 + kb);
            a0[s] = t0 * 2.0f;
            a1[s] = t1 * 2.0f;
        }
    }

    // register-resident top-K list, sorted descending by key
    float dlist[KNN];
    int   ilist[KNN];
    #pragma unroll
    for (int t = 0; t < KNN; ++t) { dlist[t] = NEG_BIG; ilist[t] = 0; }

    for (int j = 0; j < N; j += 16) {
        // this lane's column norm for the tile (col = lane&15)
        float sqj = sqb[j + (lane & 15)];
        // B fragment: rows of X transposed -> B[k][n] = X[j+n][k]
        v8f c0 = {}; v8f c1 = {};
        {
            const int n  = lane & 15;
            const int kb = (lane >> 4) * 2;
            const float* rb = Xb + (size_t)(j + n) * CPAD;
            #pragma unroll
            for (int s = 0; s < KSTEPS; ++s) {
                v2f bf = *(const v2f*)(rb + s * 4 + kb);
                c0 = __builtin_amdgcn_wmma_f32_16x16x4_f32(
                        false, a0[s], false, bf, (short)0, c0, false, false);
                c1 = __builtin_amdgcn_wmma_f32_16x16x4_f32(
                        false, a1[s], false, bf, (short)0, c1, false, false);
            }
        }
        __builtin_amdgcn_wave_barrier();
        {
            // C/D layout: VGPR r, lanes 0-15 -> M=r, lanes 16-31 -> M=8+r
            const int col   = lane & 15;
            const int rbase = (lane >> 4) * 8;
            #pragma unroll
            for (int r = 0; r < 8; ++r) {
                tile[wave][rbase + r][col]      = c0[r] - sqj;
                tile[wave][16 + rbase + r][col] = c1[r] - sqj;
            }
        }
        __builtin_amdgcn_wave_barrier();
        // selection: lane owns row (i0 + lane); key already includes -||xj||^2
        #pragma unroll 1
        for (int c = 0; c < 16; ++c) {
            float key = tile[wave][lane][c];
            if (key > dlist[KNN - 1]) {
                float dd = key; int ii = j + c;
                #pragma unroll
                for (int t = 0; t < KNN; ++t) {
                    bool  sw = dd > dlist[t];     // strict: stable tie-break
                    float od = dlist[t]; int oi = ilist[t];
                    dlist[t] = sw ? dd : od;  ilist[t] = sw ? ii : oi;
                    dd       = sw ? od : dd;  ii       = sw ? oi : ii;
                }
            }
        }
        __builtin_amdgcn_wave_barrier();
    }

    int* out = idx + ((size_t)b * N + i0 + lane) * KNN;
    #pragma unroll
    for (int t = 0; t < KNN; ++t) out[t] = ilist[t];
}

// ---------------------------------------------------------------------------
// EdgeConv: h[o] = Wa[o]·feat + (Wb-Wa)[o]·ctr ; BN + LeakyReLU ; max over k.
// One thread per point. W / BN affine staged in LDS (uniform broadcast reads).
// Optionally emits squared norms of the output (for the next kNN).
// ---------------------------------------------------------------------------
template<int CIN, int CPIN, int COUT>
__global__ __launch_bounds__(256)
void edgeconv_kernel(const float* __restrict__ X,   // (B,N,CPIN)
                     const int*   __restrict__ idx, // (B,N,KNN)
                     const float* __restrict__ W,   // (COUT, 2*CIN)
                     const float* __restrict__ g, const float* __restrict__ bt,
                     const float* __restrict__ m, const float* __restrict__ v,
                     float* __restrict__ Y,         // (B,N,COUT)
                     float* __restrict__ sqout,     // (B,N) or nullptr
                     int N)
{
    __shared__ float Wa[COUT][CIN], Wd[COUT][CIN];
    __shared__ float sc[COUT], sh[COUT];
    for (int t = threadIdx.x; t < COUT * CIN; t += 256) {
        int o = t / CIN, c = t % CIN;
        float wa = W[o * (2 * CIN) + c];
        float wb = W[o * (2 * CIN) + CIN + c];
        Wa[o][c] = wa; Wd[o][c] = wb - wa;
    }
    for (int t = threadIdx.x; t < COUT; t += 256) {
        float s = g[t] * rsqrtf(v[t] + 1e-5f);
        sc[t] = s; sh[t] = bt[t] - m[t] * s;
    }
    __syncthreads();

    const int n = blockIdx.x * 256 + threadIdx.x;
    const int b = blockIdx.y;
    const size_t p = (size_t)b * N + n;

    float ctr[CIN];
    {
        const float* xr = X + p * CPIN;
        #pragma unroll
        for (int c = 0; c < CIN; ++c) ctr[c] = xr[c];
    }
    float base[COUT];
    #pragma unroll
    for (int o = 0; o < COUT; ++o) {
        float a = 0.0f;
        #pragma unroll
        for (int c = 0; c < CIN; ++c) a += Wd[o][c] * ctr[c];
        base[o] = a;
    }
    float acc[COUT];
    #pragma unroll
    for (int o = 0; o < COUT; ++o) acc[o] = NEG_BIG;

    const int*   ip = idx + p * KNN;
    const float* Xb = X + (size_t)b * N * CPIN;
    for (int k = 0; k < KNN; ++k) {
        int j = ip[k];
        const float* fr = Xb + (size_t)j * CPIN;
        float f[CIN];
        #pragma unroll
        for (int c = 0; c < CIN; ++c) f[c] = fr[c];
        #pragma unroll
        for (int o = 0; o < COUT; ++o) {
            float h = base[o];
            #pragma unroll
            for (int c = 0; c < CIN; ++c) h += Wa[o][c] * f[c];
            h = h * sc[o] + sh[o];
            h = h > 0.0f ? h : 0.2f * h;
            acc[o] = fmaxf(acc[o], h);
        }
    }

    float* yr = Y + p * COUT;
    float s2 = 0.0f;
    #pragma unroll
    for (int o = 0; o < COUT; ++o) { yr[o] = acc[o]; s2 += acc[o] * acc[o]; }
    if (sqout) sqout[p] = s2;
}

// ---------------------------------------------------------------------------
// layer5: concat(x1,x2,x3,x4) (80ch) x W5 (32x80) + BN + LeakyReLU -> h5
// ---------------------------------------------------------------------------
__global__ __launch_bounds__(256)
void layer5_kernel(const float* __restrict__ x1, const float* __restrict__ x2,
                   const float* __restrict__ x3, const float* __restrict__ x4,
                   const float* __restrict__ W,  const float* __restrict__ g,
                   const float* __restrict__ bt, const float* __restrict__ m,
                   const float* __restrict__ v,  float* __restrict__ h5, int N)
{
    __shared__ float Wl[32][80];
    __shared__ float sc[32], sh[32];
    for (int t = threadIdx.x; t < 32 * 80; t += 256) Wl[t / 80][t % 80] = W[t];
    if (threadIdx.x < 32) {
        int o = threadIdx.x;
        float s = g[o] * rsqrtf(v[o] + 1e-5f);
        sc[o] = s; sh[o] = bt[o] - m[o] * s;
    }
    __syncthreads();

    const int n = blockIdx.x * 256 + threadIdx.x;
    const int b = blockIdx.y;
    const size_t p = (size_t)b * N + n;

    float acc[32];
    #pragma unroll
    for (int o = 0; o < 32; ++o) acc[o] = 0.0f;

    const float* r1 = x1 + p * 16;
    #pragma unroll
    for (int c = 0; c < 16; ++c) {
        float fv = r1[c];
        #pragma unroll
        for (int o = 0; o < 32; ++o) acc[o] += Wl[o][c] * fv;
    }
    const float* r2 = x2 + p * 16;
    #pragma unroll
    for (int c = 0; c < 16; ++c) {
        float fv = r2[c];
        #pragma unroll
        for (int o = 0; o < 32; ++o) acc[o] += Wl[o][16 + c] * fv;
    }
    const float* r3 = x3 + p * 16;
    #pragma unroll
    for (int c = 0; c < 16; ++c) {
        float fv = r3[c];
        #pragma unroll
        for (int o = 0; o < 32; ++o) acc[o] += Wl[o][32 + c] * fv;
    }
    const float* r4 = x4 + p * 32;
    #pragma unroll
    for (int c = 0; c < 32; ++c) {
        float fv = r4[c];
        #pragma unroll
        for (int o = 0; o < 32; ++o) acc[o] += Wl[o][48 + c] * fv;
    }

    float* yr = h5 + p * 32;
    #pragma unroll
    for (int o = 0; o < 32; ++o) {
        float a = acc[o] * sc[o] + sh[o];
        yr[o] = a > 0.0f ? a : 0.2f * a;
    }
}

// ---------------------------------------------------------------------------
// pool: per (b, channel) max + mean over N  -> pooled (B, 64) = [max | mean]
// ---------------------------------------------------------------------------
__global__ __launch_bounds__(256)
void pool_kernel(const float* __restrict__ h5, float* __restrict__ pooled, int N)
{
    const int o = blockIdx.x;       // 0..31
    const int b = blockIdx.y;
    const float* src = h5 + (size_t)b * N * 32 + o;
    float vmax = NEG_BIG, vsum = 0.0f;
    for (int n = threadIdx.x; n < N; n += 256) {
        float x = src[(size_t)n * 32];
        vmax = fmaxf(vmax, x);
        vsum += x;
    }
    __shared__ float smax[256], ssum[256];
    smax[threadIdx.x] = vmax; ssum[threadIdx.x] = vsum;
    __syncthreads();
    for (int s = 128; s > 0; s >>= 1) {
        if (threadIdx.x < (unsigned)s) {
            smax[threadIdx.x] = fmaxf(smax[threadIdx.x], smax[threadIdx.x + s]);
            ssum[threadIdx.x] += ssum[threadIdx.x + s];
        }
        __syncthreads();
    }
    if (threadIdx.x == 0) {
        pooled[b * 64 + o]      = smax[0];
        pooled[b * 64 + 32 + o] = ssum[0] / (float)N;
    }
}

// ---------------------------------------------------------------------------
// fc head: pooled(8,64) -> 32 -> 16 -> 40, single block
// ---------------------------------------------------------------------------
__global__ __launch_bounds__(320)
void fc_kernel(const float* __restrict__ pooled,
               const float* __restrict__ l1, const float* __restrict__ g6,
               const float* __restrict__ b6, const float* __restrict__ m6,
               const float* __restrict__ v6,
               const float* __restrict__ l2, const float* __restrict__ l2b,
               const float* __restrict__ g7, const float* __restrict__ b7,
               const float* __restrict__ m7, const float* __restrict__ v7,
               const float* __restrict__ l3, const float* __restrict__ l3b,
               float* __restrict__ out)
{
    __shared__ float h1[8][32], h2[8][16];
    const int t = threadIdx.x;
    if (t < 256) {
        int b = t >> 5, o = t & 31;
        float a = 0.0f;
        #pragma unroll
        for (int c = 0; c < 64; ++c) a += pooled[b * 64 + c] * l1[o * 64 + c];
        float s = g6[o] * rsqrtf(v6[o] + 1e-5f);
        a = (a - m6[o]) * s + b6[o];
        h1[b][o] = a > 0.0f ? a : 0.2f * a;
    }
    __syncthreads();
    if (t < 128) {
        int b = t >> 4, o = t & 15;
        float a = l2b[o];
        #pragma unroll
        for (int c = 0; c < 32; ++c) a += h1[b][c] * l2[o * 32 + c];
        float s = g7[o] * rsqrtf(v7[o] + 1e-5f);
        a = (a - m7[o]) * s + b7[o];
        h2[b][o] = a > 0.0f ? a : 0.2f * a;
    }
    __syncthreads();
    if (t < 320) {
        int b = t / 40, o = t % 40;
        float a = l3b[o];
        #pragma unroll
        for (int c = 0; c < 16; ++c) a += h2[b][c] * l3[o * 16 + c];
        out[b * 40 + o] = a;
    }
}

// ---------------------------------------------------------------------------
extern "C" void kernel_launch(void* const* d_in, const int* in_sizes, int n_in,
                              void* d_out, int out_size, void* d_ws, size_t ws_size,
                              hipStream_t stream)
{
    const int B = 8, N = 4096;
    (void)in_sizes; (void)n_in; (void)out_size; (void)ws_size;

    const float* x   = (const float*)d_in[0];
    const float* w1  = (const float*)d_in[1];
    const float* g1  = (const float*)d_in[2];
    const float* b1  = (const float*)d_in[3];
    const float* m1  = (const float*)d_in[4];
    const float* v1  = (const float*)d_in[5];
    const float* w2  = (const float*)d_in[6];
    const float* g2  = (const float*)d_in[7];
    const float* b2  = (const float*)d_in[8];
    const float* m2  = (const float*)d_in[9];
    const float* v2  = (const float*)d_in[10];
    const float* w3  = (const float*)d_in[11];
    const float* g3  = (const float*)d_in[12];
    const float* b3  = (const float*)d_in[13];
    const float* m3  = (const float*)d_in[14];
    const float* v3  = (const float*)d_in[15];
    const float* w4  = (const float*)d_in[16];
    const float* g4  = (const float*)d_in[17];
    const float* b4  = (const float*)d_in[18];
    const float* m4  = (const float*)d_in[19];
    const float* v4  = (const float*)d_in[20];
    const float* w5  = (const float*)d_in[21];
    const float* g5  = (const float*)d_in[22];
    const float* b5  = (const float*)d_in[23];
    const float* m5  = (const float*)d_in[24];
    const float* v5  = (const float*)d_in[25];
    const float* l1  = (const float*)d_in[26];
    const float* g6  = (const float*)d_in[27];
    const float* b6  = (const float*)d_in[28];
    const float* m6  = (const float*)d_in[29];
    const float* v6  = (const float*)d_in[30];
    const float* l2  = (const float*)d_in[31];
    const float* l2b = (const float*)d_in[32];
    const float* g7  = (const float*)d_in[33];
    const float* b7  = (const float*)d_in[34];
    const float* m7  = (const float*)d_in[35];
    const float* v7  = (const float*)d_in[36];
    const float* l3  = (const float*)d_in[37];
    const float* l3b = (const float*)d_in[38];

    // workspace carve-out (floats); totals ~18.0 MB
    float* ws     = (float*)d_ws;
    float* xt     = ws;                         // B*N*4   = 131072
    float* sqv    = xt + (size_t)B * N * 4;     // B*N     = 32768
    int*   idxb   = (int*)(sqv + (size_t)B * N);// B*N*20  = 655360 ints
    float* x1     = (float*)(idxb + (size_t)B * N * KNN); // B*N*16
    float* x2     = x1 + (size_t)B * N * 16;
    float* x3     = x2 + (size_t)B * N * 16;
    float* x4     = x3 + (size_t)B * N * 16;    // B*N*32
    float* h5     = x4 + (size_t)B * N * 32;    // B*N*32
    float* pooled = h5 + (size_t)B * N * 32;    // B*64

    dim3 gpt(N / 256, B);     // per-point kernels
    dim3 gkn(N / 128, B);     // knn: 4 waves x 32 rows per block

    prep_kernel<<<gpt, 256, 0, stream>>>(x, xt, sqv, N);

    // layer 1: knn on 3D coords (padded to 4)
    knn_kernel<4><<<gkn, 128, 0, stream>>>(xt, sqv, idxb, N);
    edgeconv_kernel<3, 4, 16><<<gpt, 256, 0, stream>>>(
        xt, idxb, w1, g1, b1, m1, v1, x1, sqv, N);

    // layer 2
    knn_kernel<16><<<gkn, 128, 0, stream>>>(x1, sqv, idxb, N);
    edgeconv_kernel<16, 16, 16><<<gpt, 256, 0, stream>>>(
        x1, idxb, w2, g2, b2, m2, v2, x2, sqv, N);

    // layer 3
    knn_kernel<16><<<gkn, 128, 0, stream>>>(x2, sqv, idxb, N);
    edgeconv_kernel<16, 16, 16><<<gpt, 256, 0, stream>>>(
        x2, idxb, w3, g3, b3, m3, v3, x3, sqv, N);

    // layer 4
    knn_kernel<16><<<gkn, 128, 0, stream>>>(x3, sqv, idxb, N);
    edgeconv_kernel<16, 16, 32><<<gpt, 256, 0, stream>>>(
        x3, idxb, w4, g4, b4, m4, v4, x4, nullptr, N);

    // layer 5 + pooling + head
    layer5_kernel<<<gpt, 256, 0, stream>>>(x1, x2, x3, x4, w5, g5, b5, m5, v5, h5, N);
    pool_kernel<<<dim3(32, B), 256, 0, stream>>>(h5, pooled, N);
    fc_kernel<<<1, 320, 0, stream>>>(pooled, l1, g6, b6, m6, v6,
                                     l2, l2b, g7, b7, m7, v7, l3, l3b,
                                     (float*)d_out);
}